// KNNSegmentator_43207370998078
// MI455X (gfx1250) — compile-verified
//
#include <hip/hip_runtime.h>
#include <hip/hip_bf16.h>
#include <math.h>

// Problem constants (match reference)
#define BB 16     // batch
#define PP 196    // patches
#define DD 768    // feature dim
#define NN 2048   // train bank
#define KK 20     // top-k
#define PSQ 256   // PS*PS
#define NR 14     // sqrt(P)
#define IMG 224   // NR*PS

typedef __attribute__((ext_vector_type(2))) float v2f;
typedef __attribute__((ext_vector_type(8))) float v8f;

// ---------------------------------------------------------------------------
// Kernel 1: per-patch GEMM  sim[b,p,n] = sum_d test[b,p,d] * train[p,d,n]
// One wave handles one patch x 64-column strip using V_WMMA_F32_16X16X4_F32.
// M = B = 16 rows exactly fills the WMMA tile.
// blockDim = 256 (8 waves). 4 blocks per patch (8 waves * 64 cols = 512;
// 4 blocks -> 2048 cols). grid = P * 4 = 784 blocks.
// train_features is streamed exactly once (1.23 GB -> ~53us @ 23.3 TB/s).
// ---------------------------------------------------------------------------
__global__ __launch_bounds__(256) void sim_wmma_kernel(
    const float* __restrict__ test,   // [B, P, D]
    const float* __restrict__ train,  // [P, D, N]
    float* __restrict__ sim)          // [B, P, N] (workspace)
{
    const int lane = threadIdx.x & 31;
    const int wave = threadIdx.x >> 5;
    const int p    = blockIdx.x >> 2;                 // patch
    const int wip  = ((blockIdx.x & 3) << 3) | wave;  // 0..31 strip id
    const int n0   = wip << 6;                        // strip base column (64 wide)
    const int half = lane >> 4;                       // 0: lanes 0-15, 1: lanes 16-31
    const int l16  = lane & 15;

    // A-matrix: row M = l16 (batch index), K pair selected by half-wave
    const float* __restrict__ arow = test + (size_t)l16 * (PP * DD) + (size_t)p * DD;
    // B-matrix: train tile for this patch
    const float* __restrict__ bmat = train + (size_t)p * (DD * NN);

    v8f acc0 = {}, acc1 = {}, acc2 = {}, acc3 = {};

    for (int d = 0; d < DD; d += 4) {
        const int kk = d + (half << 1); // K index pair base for this half-wave
        v2f a;
        a.x = arow[kk];
        a.y = arow[kk + 1];

        const float* __restrict__ brow0 = bmat + (size_t)kk * NN + n0 + l16;
        const float* __restrict__ brow1 = brow0 + NN;

        v2f b0, b1, b2, b3;
        b0.x = brow0[0];   b0.y = brow1[0];
        b1.x = brow0[16];  b1.y = brow1[16];
        b2.x = brow0[32];  b2.y = brow1[32];
        b3.x = brow0[48];  b3.y = brow1[48];

        acc0 = __builtin_amdgcn_wmma_f32_16x16x4_f32(false, a, false, b0, (short)0, acc0, false, false);
        acc1 = __builtin_amdgcn_wmma_f32_16x16x4_f32(false, a, false, b1, (short)0, acc1, false, false);
        acc2 = __builtin_amdgcn_wmma_f32_16x16x4_f32(false, a, false, b2, (short)0, acc2, false, false);
        acc3 = __builtin_amdgcn_wmma_f32_16x16x4_f32(false, a, false, b3, (short)0, acc3, false, false);
    }

    // C/D layout: VGPR r, lanes 0-15 -> row r, lanes 16-31 -> row r+8; col = l16.
    v8f accs[4] = {acc0, acc1, acc2, acc3};
#pragma unroll
    for (int s = 0; s < 4; ++s) {
        const int n = n0 + (s << 4) + l16;
        float* __restrict__ out = sim + (size_t)p * NN + n;
#pragma unroll
        for (int r = 0; r < 8; ++r) {
            const int m = r + (half << 3); // batch row
            out[(size_t)m * (PP * NN)] = accs[s][r];
        }
    }
}

// ---------------------------------------------------------------------------
// Kernel 2: top-K (sorted, descending, ties -> smaller index) per (b,p) row.
// One workgroup per row; row cached in LDS; 20 extract-max passes.
// ---------------------------------------------------------------------------
__global__ __launch_bounds__(256) void topk_kernel(
    const float* __restrict__ sim,  // [B, P, N]
    float* __restrict__ dist,       // [B, P, K]
    int* __restrict__ topidx)       // [B, P, K]
{
    const int row = blockIdx.x; // b*P + p
    const float* __restrict__ srow = sim + (size_t)row * NN;

    __shared__ float vals[NN];
    __shared__ float sval[256];
    __shared__ int   sidx[256];

    for (int i = threadIdx.x; i < NN; i += 256) vals[i] = srow[i];
    __syncthreads();

    for (int k = 0; k < KK; ++k) {
        float bestv = -INFINITY;
        int   besti = 0x7fffffff;
        for (int i = threadIdx.x; i < NN; i += 256) {
            const float v = vals[i];
            if (v > bestv || (v == bestv && i < besti)) { bestv = v; besti = i; }
        }
        sval[threadIdx.x] = bestv;
        sidx[threadIdx.x] = besti;
        __syncthreads();
#pragma unroll
        for (int off = 128; off > 0; off >>= 1) {
            if ((int)threadIdx.x < off) {
                const float v2 = sval[threadIdx.x + off];
                const int   i2 = sidx[threadIdx.x + off];
                if (v2 > sval[threadIdx.x] ||
                    (v2 == sval[threadIdx.x] && i2 < sidx[threadIdx.x])) {
                    sval[threadIdx.x] = v2;
                    sidx[threadIdx.x] = i2;
                }
            }
            __syncthreads();
        }
        if (threadIdx.x == 0) {
            dist[(size_t)row * KK + k]   = sval[0];
            topidx[(size_t)row * KK + k] = sidx[0];
            vals[sidx[0]] = -INFINITY;  // remove winner for next pass
        }
        __syncthreads();
    }
}

// ---------------------------------------------------------------------------
// Kernel 3: label gather + grid assembly.
// grid[b,k, pr*16+si, pc*16+sj] = labels[p, si*16+sj, topidx[b,p,k]],
// p = pr*14 + pc. Written as float values of the int labels.
// ---------------------------------------------------------------------------
__global__ __launch_bounds__(256) void grid_kernel(
    const int* __restrict__ labels,   // [P, 256, N]
    const int* __restrict__ topidx,   // [B, P, K]
    float* __restrict__ grid)         // [B, K, 224, 224]
{
    const size_t total = (size_t)BB * KK * IMG * IMG;
    const size_t t = (size_t)blockIdx.x * 256 + threadIdx.x;
    if (t >= total) return;

    const int col = (int)(t % IMG);
    size_t tmp = t / IMG;
    const int row = (int)(tmp % IMG);
    tmp /= IMG;
    const int k = (int)(tmp % KK);
    const int b = (int)(tmp / KK);

    const int pr = row >> 4, si = row & 15;
    const int pc = col >> 4, sj = col & 15;
    const int p  = pr * NR + pc;
    const int s  = (si << 4) | sj;

    const int n   = topidx[((size_t)b * PP + p) * KK + k];
    const int lab = labels[((size_t)p * PSQ + s) * NN + n];
    grid[t] = (float)lab;
}

// ---------------------------------------------------------------------------
// Launch
// ---------------------------------------------------------------------------
extern "C" void kernel_launch(void* const* d_in, const int* in_sizes, int n_in,
                              void* d_out, int out_size, void* d_ws, size_t ws_size,
                              hipStream_t stream) {
    (void)in_sizes; (void)n_in; (void)out_size; (void)ws_size;

    const float* test   = (const float*)d_in[0]; // [B, P, D]
    const float* train  = (const float*)d_in[1]; // [P, D, N]
    const int*   labels = (const int*)d_in[2];   // [P, 256, N]

    // Workspace layout: sim [B*P*N] f32, then topidx [B*P*K] i32.
    float* sim    = (float*)d_ws;
    int*   topidx = (int*)(sim + (size_t)BB * PP * NN);

    // Output layout: dist [B*P*K] f32, then grid [B*K*224*224].
    float* dist = (float*)d_out;
    float* grid = dist + (size_t)BB * PP * KK;

    // 1) GEMM: 4 blocks/patch * 196 patches, 256 threads (8 waves) each.
    sim_wmma_kernel<<<PP * 4, 256, 0, stream>>>(test, train, sim);

    // 2) top-K: one block per (b,p) row.
    topk_kernel<<<BB * PP, 256, 0, stream>>>(sim, dist, topidx);

    // 3) gather + grid assembly.
    const size_t total = (size_t)BB * KK * IMG * IMG;
    const int blocks = (int)((total + 255) / 256);
    grid_kernel<<<blocks, 256, 0, stream>>>(labels, topidx, grid);
}